// ClusteredLogSoftmax_70454643523922
// MI455X (gfx1250) — compile-verified
//
#include <hip/hip_runtime.h>

// Clustered log-softmax, MI455X (gfx1250).
// Bandwidth-bound: ~1.2 GB HBM traffic -> ~52us floor @ 23.3 TB/s.
// One block per row; native ds_add_f32 segmented reduction (workgroup-scope
// relaxed fp atomics); double-buffered global_load_async_to_lds_b128 staging
// (ASYNCcnt) for the pass-1 stream; non-temporal b128 output stores so the
// logits row stays L2-resident for the pass-2 re-read.

#define BLOCK 256
#define MAX_C 1024
#define CHUNK 1024 // floats per staged chunk = BLOCK lanes * 4 floats (b128)

typedef int v4i __attribute__((vector_size(16)));
typedef float v4f __attribute__((vector_size(16)));

#if defined(__HIP_DEVICE_COMPILE__) && defined(__gfx1250__) && \
    __has_builtin(__builtin_amdgcn_global_load_async_to_lds_b128)
#define USE_ASYNC_LDS 1
#else
#define USE_ASYNC_LDS 0
#endif

#if defined(__HIP_DEVICE_COMPILE__) && __has_builtin(__builtin_amdgcn_s_wait_asynccnt)
#define WAIT_ASYNC(n) __builtin_amdgcn_s_wait_asynccnt(n)
#else
#define WAIT_ASYNC(n) asm volatile("s_wait_asynccnt %0" ::"i"(n) : "memory")
#endif

#if USE_ASYNC_LDS
__device__ __forceinline__ void async_copy_b128(const void* gsrc, void* ldst) {
  __builtin_amdgcn_global_load_async_to_lds_b128(
      (v4i*)gsrc, (v4i*)ldst, /*offset=*/0, /*cpol=*/0);
}
#endif

// Workgroup-scope relaxed fp add on LDS -> native no-return ds_add_f32.
__device__ __forceinline__ void lds_fadd(float* addr, float v) {
  __hip_atomic_fetch_add(addr, v, __ATOMIC_RELAXED, __HIP_MEMORY_SCOPE_WORKGROUP);
}

__device__ __forceinline__ float log_sigmoid_f(float x) {
  // log_sigmoid(x) = -softplus(-x), numerically stable
  float z = __logf(1.0f + __expf(-fabsf(x)));
  return (x >= 0.0f) ? -z : (x - z);
}

extern "C" __global__ void __launch_bounds__(BLOCK)
clustered_logsoftmax_kernel(const float* __restrict__ logits,
                            const int* __restrict__ cluster,
                            const int* __restrict__ ncl,
                            float* __restrict__ out, int V) {
  __shared__ float s_sums[MAX_C];
#if USE_ASYNC_LDS
  __shared__ float s_logit[2][CHUNK];
  __shared__ int s_clus[2][CHUNK];
#endif

  const int tid = threadIdx.x;
  const long long b = blockIdx.x;
  const int C = (ncl[0] < MAX_C) ? ncl[0] : MAX_C;

  const float* __restrict__ row = logits + b * (long long)V;
  float* __restrict__ orow = out + b * (long long)V;

  for (int c = tid; c < MAX_C; c += BLOCK) s_sums[c] = 0.0f;
  __syncthreads();

  const bool vec_ok = ((V & 3) == 0); // rows 16B-aligned, b128 legal
  const int nchunk = vec_ok ? (V / CHUNK) : 0;
  const int nv4 = vec_ok ? (V >> 2) : 0;

  // ---------------- Pass 1: segment sums of exp over clusters ----------------
#if USE_ASYNC_LDS
  if (nchunk > 0) { // prime chunk 0
    async_copy_b128(row + tid * 4, &s_logit[0][tid * 4]);
    async_copy_b128(cluster + tid * 4, &s_clus[0][tid * 4]);
  }
  for (int ci = 0; ci < nchunk; ++ci) {
    const int cur = ci & 1;
    if (ci + 1 < nchunk) { // issue next chunk, then wait only for current
      const long long nb = (long long)(ci + 1) * CHUNK;
      async_copy_b128(row + nb + tid * 4, &s_logit[1 - cur][tid * 4]);
      async_copy_b128(cluster + nb + tid * 4, &s_clus[1 - cur][tid * 4]);
      WAIT_ASYNC(2);
    } else {
      WAIT_ASYNC(0);
    }
    __syncthreads();
#pragma unroll
    for (int k = 0; k < CHUNK / BLOCK; ++k) {
      const int i = tid + k * BLOCK;
      lds_fadd(&s_sums[s_clus[cur][i]], __expf(s_logit[cur][i]));
    }
    __syncthreads(); // buffer 'cur' free for reuse next iteration
  }
  const int tail0 = nchunk * CHUNK;
#else
  {
    const v4f* row4 = (const v4f*)row;
    const v4i* cl4 = (const v4i*)cluster;
    for (int i = tid; i < nv4; i += BLOCK) {
      v4f x = row4[i];
      v4i c = cl4[i];
#pragma unroll
      for (int j = 0; j < 4; ++j) lds_fadd(&s_sums[c[j]], __expf(x[j]));
    }
  }
  const int tail0 = nv4 << 2;
#endif
  for (int v = tail0 + tid; v < V; v += BLOCK)
    lds_fadd(&s_sums[cluster[v]], __expf(row[v]));
  __syncthreads();

  // ------------- log(clamp(sums)) in place -------------
  for (int c = tid; c < C; c += BLOCK)
    s_sums[c] = __logf(fmaxf(s_sums[c], 1e-20f));
  __syncthreads();

  // ---------------- Pass 2: emit outputs (NT stores) ----------------
  {
    const v4f* row4 = (const v4f*)row;
    const v4i* cl4 = (const v4i*)cluster;
    v4f* o4 = (v4f*)orow;
    for (int i = tid; i < nv4; i += BLOCK) {
      v4f x = row4[i];
      v4i c = cl4[i];
      v4f r;
#pragma unroll
      for (int j = 0; j < 4; ++j)
        r[j] = (c[j] == 0) ? log_sigmoid_f(x[j]) : x[j] - s_sums[c[j]];
      __builtin_nontemporal_store(r, &o4[i]);
    }
    for (int v = (nv4 << 2) + tid; v < V; v += BLOCK) {
      float x = row[v];
      int c = cluster[v];
      float r = (c == 0) ? log_sigmoid_f(x) : x - s_sums[c];
      __builtin_nontemporal_store(r, &orow[v]);
    }
  }
}

extern "C" void kernel_launch(void* const* d_in, const int* in_sizes, int n_in,
                              void* d_out, int out_size, void* d_ws,
                              size_t ws_size, hipStream_t stream) {
  (void)n_in; (void)out_size; (void)d_ws; (void)ws_size;
  const float* logits = (const float*)d_in[0];
  const int* cluster = (const int*)d_in[1];
  const int* ncl = (const int*)d_in[2];
  float* out = (float*)d_out;

  const int V = in_sizes[1];
  const int B = in_sizes[0] / V;

  hipLaunchKernelGGL(clustered_logsoftmax_kernel, dim3(B), dim3(BLOCK), 0,
                     stream, logits, cluster, ncl, out, V);
}